// HierarchicalSoftmax_32169305047407
// MI455X (gfx1250) — compile-verified
//
#include <hip/hip_runtime.h>

// Problem constants (match reference)
#define V_SIZE  131072
#define DEPTH   17
#define BATCH   16384
#define EDIM    128
#define TILES   (BATCH / 16)          // 1024 tiles of 16 samples

typedef __attribute__((ext_vector_type(2))) float v2f;
typedef __attribute__((ext_vector_type(8))) float v8f;

// One wave (one single-wave workgroup) handles 16 samples across all 17 tree
// levels. A (16x128 f32 embeddings) stays resident in 64 VGPRs. B gathers are
// double-buffered across levels so ~64 loads stay in flight; 32x
// v_wmma_f32_16x16x4_f32 per level accumulate into two C registers.
__global__ __launch_bounds__(32) void hs_wmma_kernel(
    const float* __restrict__ emb,
    const int*   __restrict__ word_idx,
    const float* __restrict__ W,
    const float* __restrict__ bias,
    float*       __restrict__ partials)
{
    __shared__ float diagbuf[32 * 8];
    __shared__ float wsum[16];

    const int lane = threadIdx.x & 31;
    const int tile = blockIdx.x;
    const int row  = lane & 15;       // sample-within-tile == matrix row/col
    const int half = lane >> 4;       // 0: K%4 in {0,1}, 1: K%4 in {2,3}
    const int sample = tile * 16 + row;

    // ---- A fragments: 16 samples x 128 f32, resident for all 17 levels ----
    v2f a[32];
    const v2f* erow = reinterpret_cast<const v2f*>(emb + (size_t)sample * EDIM);
    #pragma unroll
    for (int k = 0; k < 32; ++k) a[k] = erow[2 * k + half];

    // Node chain (lanes L and L+16 compute identically for row = lane&15).
    int child = word_idx[sample] + (V_SIZE - 1);
    int node  = (child - 1) >> 1;     // node for depth 0

    // ---- Double-buffered B fragments: preload level 0 ----
    v2f bfrag[2][32];
    {
        const v2f* wrow = reinterpret_cast<const v2f*>(W + (size_t)node * EDIM);
        #pragma unroll
        for (int k = 0; k < 32; ++k) bfrag[0][k] = wrow[2 * k + half];
    }

    float acc = 0.0f;

    // Full unroll: keeps the bfrag[d&1] index a compile-time constant so both
    // buffers live in registers (dynamic indexing would force scratch).
    #pragma unroll
    for (int d = 0; d < DEPTH; ++d) {
        const int   buf = d & 1;
        const float tgt = (float)(child & 1);   // 1 iff came from left child
        const int   next_node = (node - 1) >> 1;

        // Issue next level's 32 gathers NOW so they overlap this level's WMMAs.
        if (d + 1 < DEPTH) {
            const v2f* wrow =
                reinterpret_cast<const v2f*>(W + (size_t)next_node * EDIM);
            #pragma unroll
            for (int k = 0; k < 32; ++k) bfrag[buf ^ 1][k] = wrow[2 * k + half];
            if (d + 2 < DEPTH)
                __builtin_prefetch(
                    W + (size_t)((next_node - 1) >> 1) * EDIM + 4 * half, 0, 3);
        }

        // ---- 32x WMMA f32 16x16x4 over K=128, two accumulation chains ----
        v8f c0 = {}, c1 = {};
        #pragma unroll
        for (int k = 0; k < 32; k += 2) {
            c0 = __builtin_amdgcn_wmma_f32_16x16x4_f32(
                     false, a[k],     false, bfrag[buf][k],
                     (short)0, c0, false, false);
            c1 = __builtin_amdgcn_wmma_f32_16x16x4_f32(
                     false, a[k + 1], false, bfrag[buf][k + 1],
                     (short)0, c1, false, false);
        }

        // ---- Diagonal extraction via LDS (single-wave WG: barrier == S_NOP) ----
        #pragma unroll
        for (int r = 0; r < 8; ++r) diagbuf[lane * 8 + r] = c0[r] + c1[r];
        __syncthreads();

        if (lane < 16) {
            const int j = lane;
            // C layout: VGPR r, lanes 0-15 -> (M=r, N=lane); lanes 16-31 -> (M=r+8, N=lane-16)
            float s = (j < 8) ? diagbuf[j * 8 + j]
                              : diagbuf[(j + 16) * 8 + (j - 8)];
            s += bias[node];
            // numerically-stable BCE with logits
            acc += fmaxf(s, 0.0f) - s * tgt + log1pf(expf(-fabsf(s)));
        }
        __syncthreads();

        child = node;
        node  = next_node;
    }

    // ---- Per-tile deterministic partial ----
    if (lane < 16) wsum[lane] = acc;
    __syncthreads();
    if (lane == 0) {
        float t = 0.0f;
        #pragma unroll
        for (int i = 0; i < 16; ++i) t += wsum[i];
        partials[tile] = t;
    }
}

// Deterministic single-block reduction of the 1024 tile partials.
__global__ __launch_bounds__(256) void hs_reduce_kernel(
    const float* __restrict__ partials, float* __restrict__ out)
{
    __shared__ float buf[256];
    const int t = threadIdx.x;
    float s = 0.0f;
    #pragma unroll
    for (int i = 0; i < TILES / 256; ++i) s += partials[t + i * 256];
    buf[t] = s;
    __syncthreads();
    #pragma unroll
    for (int stride = 128; stride > 0; stride >>= 1) {
        if (t < stride) buf[t] += buf[t + stride];
        __syncthreads();
    }
    if (t == 0) out[0] = buf[0] / (float)BATCH;
}

extern "C" void kernel_launch(void* const* d_in, const int* in_sizes, int n_in,
                              void* d_out, int out_size, void* d_ws, size_t ws_size,
                              hipStream_t stream)
{
    const float* emb      = (const float*)d_in[0];   // [B, E] f32
    const int*   word_idx = (const int*)  d_in[1];   // [B]    i32
    const float* W        = (const float*)d_in[2];   // [V-1, E] f32
    const float* bias     = (const float*)d_in[3];   // [V-1]  f32
    float* out            = (float*)d_out;           // scalar f32
    float* partials       = (float*)d_ws;            // 1024 f32 in scratch

    hs_wmma_kernel<<<TILES, 32, 0, stream>>>(emb, word_idx, W, bias, partials);
    hs_reduce_kernel<<<1, 256, 0, stream>>>(partials, out);
}